// CG_sparse_py_3813930959605
// MI455X (gfx1250) — compile-verified
//
#include <hip/hip_runtime.h>

// ---------------------------------------------------------------------------
// CGNet layer for MI455X (gfx1250, wave32).
//   out[b, base_l + (tupleInL*1024 + t1*32+t2)*(2l+1) + m, {re,im}]
//     = sum_k C[m,k] * kron[k, t1*32+t2]   (complex, CG matrix is real)
// GEMM per tuple: M=2l+1 (pad 16), K=m1*m2 (pad 4), N=32*1024 -> WMMA f32 16x16x4
// Inner loop is branchless: K-padding rows index a zero row in LDS.
// Each wave processes two N-tiles sharing t1 -> 4 WMMAs per A-fragment load.
// ---------------------------------------------------------------------------

#define NTUP       111
#define TOTAL_IN   2304
#define OUT_AX1    789504     // 1024 * sum_tuples (2l+1) = 1024*771
#define MAX_KSTEPS 31         // ceil(121/4)
#define ZROW       (11 * 32)  // v2f index of the zero row in aL/bL

typedef float v2f __attribute__((ext_vector_type(2)));
typedef float v8f __attribute__((ext_vector_type(8)));
typedef int   v2i __attribute__((ext_vector_type(2)));

// ---------------- tuple metadata (uniform scalar decode, matches LTUPLES) ---
struct TupInfo {
  int l, l1, l2;     // degrees
  int idxInL;        // appearance order among tuples with same output l
  int cgOff;         // float offset of packed CG A-fragments in workspace
  int ksteps;        // ceil(m1*m2 / 4)
};

__device__ __forceinline__ void decode_tuple(int t, TupInfo& ti) {
  int cnt = 0, off = 0;
  for (int L = 0; L <= 5; ++L) {
    int inL = 0;
    for (int A = 0; A <= 5; ++A) {
      for (int Bb = 0; Bb <= 5; ++Bb) {
        int d = (A > Bb) ? (A - Bb) : (Bb - A);
        if (d <= L && L <= A + Bb) {
          int Kt = (2 * A + 1) * (2 * Bb + 1);
          int ks = (Kt + 3) >> 2;
          if (cnt == t) {
            ti.l = L; ti.l1 = A; ti.l2 = Bb;
            ti.idxInL = inL; ti.cgOff = off; ti.ksteps = ks;
          }
          ++cnt; ++inL; off += ks * 64;
        }
      }
    }
  }
}

// ---------------- Clebsch-Gordan (Racah), f64: factorials <= 16! exact ------
__device__ __forceinline__ double dfact(int n) {
  double r = 1.0;
  for (int i = 2; i <= n; ++i) r *= (double)i;
  return r;
}

__device__ double cg_coef(int j1, int m1, int j2, int m2, int j3, int m3) {
  if (m1 + m2 != m3) return 0.0;
  int dj = (j1 > j2) ? (j1 - j2) : (j2 - j1);
  if (j3 < dj || j3 > j1 + j2) return 0.0;
  double pref = sqrt((double)(2 * j3 + 1) * dfact(j3 + j1 - j2) * dfact(j3 - j1 + j2)
                     * dfact(j1 + j2 - j3) / dfact(j1 + j2 + j3 + 1));
  pref *= sqrt(dfact(j3 + m3) * dfact(j3 - m3) * dfact(j1 - m1) * dfact(j1 + m1)
               * dfact(j2 - m2) * dfact(j2 + m2));
  double s = 0.0;
  for (int k = 0; k <= j1 + j2 - j3; ++k) {
    int a1 = j1 + j2 - j3 - k, a2 = j1 - m1 - k, a3 = j2 + m2 - k;
    int a4 = j3 - j2 + m1 + k, a5 = j3 - j1 - m2 + k;
    if (a1 < 0 || a2 < 0 || a3 < 0 || a4 < 0 || a5 < 0) continue;
    double den = dfact(k) * dfact(a1) * dfact(a2) * dfact(a3) * dfact(a4) * dfact(a5);
    s += ((k & 1) ? -1.0 : 1.0) / den;
  }
  return pref * s;
}

// ---------------- init: pack CG matrices into WMMA 16x4 A-fragment layout ---
// cgA[cgOff + ks*64 + lane*2 + j]:
//   lane<16 : (M=lane,    K=4*ks + j)       lane>=16: (M=lane-16, K=4*ks+2+j)
__global__ void cg_init_kernel(float* __restrict__ cgA) {
  TupInfo ti; decode_tuple((int)blockIdx.x, ti);
  int d2 = 2 * ti.l2 + 1;
  int Ktot = (2 * ti.l1 + 1) * d2;
  int Mrows = 2 * ti.l + 1;
  int nElem = ti.ksteps * 64;
  for (int e = threadIdx.x; e < nElem; e += blockDim.x) {
    int ks = e >> 6, r = e & 63, lane = r >> 1, j = r & 1;
    int mRow = lane & 15;
    int k = ks * 4 + ((lane >= 16) ? 2 : 0) + j;
    float val = 0.f;
    if (mRow < Mrows && k < Ktot) {
      int m1 = k / d2, m2 = k - m1 * d2;
      val = (float)cg_coef(ti.l1, m1 - ti.l1, ti.l2, m2 - ti.l2, ti.l, mRow - ti.l);
    }
    cgA[ti.cgOff + e] = val;
  }
}

// ---------------- main compute kernel --------------------------------------
__global__ void __launch_bounds__(256)
cg_wmma_kernel(const float* __restrict__ fs, float* __restrict__ out,
               const float* __restrict__ cgA) {
  // LDS staging (~15 KB of 320 KB/WGP); row 11 of aL/bL is an explicit zero row
  __shared__ __align__(16) float aL[12 * 64];               // [m1][tau][re,im]
  __shared__ __align__(16) float bL[12 * 64];               // [m2][tau][re,im]
  __shared__ __align__(16) float cgL[MAX_KSTEPS * 64];      // packed A-frags
  __shared__ __align__(16) int   kOff[MAX_KSTEPS * 4];      // k -> (m1*32 | m2*32<<16)

  const int t = (int)blockIdx.x >> 5;     // tuple id
  const int b = (int)blockIdx.x & 31;     // batch id
  TupInfo ti; decode_tuple(t, ti);

  const int OFFL[6]  = {0, 64, 256, 576, 1024, 1600};                 // fs offsets
  const int BASEL[6] = {0, 6144, 52224, 159744, 331776, 552960};      // axis-1 bases

  const int d1 = 2 * ti.l1 + 1, d2 = 2 * ti.l2 + 1;
  const int Ktot = d1 * d2, Mrows = 2 * ti.l + 1;

  const float* __restrict__ fsrow = fs + b * TOTAL_IN;
  __builtin_prefetch(fsrow, 0, 1);   // global_prefetch_b8

  // cooperative staging:  a[m][tau][c] = fs[off + (tau*d + m)*2 + c]
  for (int e = threadIdx.x; e < d1 * 64; e += 256) {
    int m = e >> 6, r = e & 63, tau = r >> 1, c = r & 1;
    aL[e] = fsrow[OFFL[ti.l1] + (tau * d1 + m) * 2 + c];
  }
  for (int e = threadIdx.x; e < d2 * 64; e += 256) {
    int m = e >> 6, r = e & 63, tau = r >> 1, c = r & 1;
    bL[e] = fsrow[OFFL[ti.l2] + (tau * d2 + m) * 2 + c];
  }
  // zero row for branchless K-padding
  for (int e = threadIdx.x; e < 64; e += 256) {
    aL[11 * 64 + e] = 0.f;
    bL[11 * 64 + e] = 0.f;
  }
  for (int e = threadIdx.x; e < ti.ksteps * 64; e += 256)
    cgL[e] = cgA[ti.cgOff + e];
  for (int k = threadIdx.x; k < ti.ksteps * 4; k += 256) {
    int off = ZROW | (ZROW << 16);                 // pad rows hit the zero row
    if (k < Ktot) {
      int m1 = k / d2, m2 = k - m1 * d2;
      off = (m1 * 32) | ((m2 * 32) << 16);
    }
    kOff[k] = off;
  }
  __syncthreads();

  const int lane = threadIdx.x & 31;
  const int wave = (int)threadIdx.x >> 5;
  const int ln   = lane & 15;
  const bool hi  = lane >= 16;

  const v2f* aLv   = (const v2f*)aL;
  const v2f* bLv   = (const v2f*)bL;
  const v2f* cgLv  = (const v2f*)cgL;
  const v2i* kOffv = (const v2i*)kOff;
  v2f* out2 = (v2f*)out;
  const int outBase = b * OUT_AX1 + BASEL[ti.l] + ti.idxInL * 1024 * Mrows;
  const int ksteps = ti.ksteps;

  // 32 tile-pairs (two 16-col tiles sharing t1); 8 waves round-robin
  for (int pair = wave; pair < 32; pair += 8) {
    const int t1 = pair;                 // both tiles in the pair share t1
    v8f accR0 = {}; v8f accI0 = {};      // tile A: columns t2 = 0..15  (+ln)
    v8f accR1 = {}; v8f accI1 = {};      // tile B: columns t2 = 16..31 (+ln)
    for (int ks = 0; ks < ksteps; ++ks) {
      v2f aF = cgLv[ks * 32 + lane];                 // CG A-fragment (pre-packed)
      const int k0 = ks * 4 + (hi ? 2 : 0);          // even -> aligned v2i load
      v2i ko = kOffv[k0 >> 1];                       // offsets for rows k0, k0+1
      // shared left operands (same t1 for both tiles)
      v2f av0 = aLv[(ko.x & 0xffff) + t1];
      v2f av1 = aLv[(ko.y & 0xffff) + t1];
      // per-tile right operands
      v2f bv0 = bLv[(ko.x >> 16) + ln];
      v2f bv1 = bLv[(ko.y >> 16) + ln];
      v2f cv0 = bLv[(ko.x >> 16) + ln + 16];
      v2f cv1 = bLv[(ko.y >> 16) + ln + 16];
      // complex kron rows (branchless; pad rows are zeros)
      v2f bR, bI, cR, cI;
      bR.x = av0.x * bv0.x - av0.y * bv0.y;  bI.x = av0.x * bv0.y + av0.y * bv0.x;
      bR.y = av1.x * bv1.x - av1.y * bv1.y;  bI.y = av1.x * bv1.y + av1.y * bv1.x;
      cR.x = av0.x * cv0.x - av0.y * cv0.y;  cI.x = av0.x * cv0.y + av0.y * cv0.x;
      cR.y = av1.x * cv1.x - av1.y * cv1.y;  cI.y = av1.x * cv1.y + av1.y * cv1.x;
      // 4 independent accumulator chains on the f32 matrix pipe
      accR0 = __builtin_amdgcn_wmma_f32_16x16x4_f32(false, aF, false, bR,
                                                    (short)0, accR0, false, false);
      accI0 = __builtin_amdgcn_wmma_f32_16x16x4_f32(false, aF, false, bI,
                                                    (short)0, accI0, false, false);
      accR1 = __builtin_amdgcn_wmma_f32_16x16x4_f32(false, aF, false, cR,
                                                    (short)0, accR1, false, false);
      accI1 = __builtin_amdgcn_wmma_f32_16x16x4_f32(false, aF, false, cI,
                                                    (short)0, accI1, false, false);
    }
    // D layout: VGPR r -> (M = r + (hi?8:0), N = ln). Pack re/im -> b64 stores.
    const int colA = outBase + (pair * 32 + ln) * Mrows;
    const int colB = colA + 16 * Mrows;
#pragma unroll
    for (int rr = 0; rr < 8; ++rr) {
      int m = rr + (hi ? 8 : 0);
      if (m < Mrows) {
        v2f vA; vA.x = accR0[rr]; vA.y = accI0[rr];
        v2f vB; vB.x = accR1[rr]; vB.y = accI1[rr];
        out2[colA + m] = vA;
        out2[colB + m] = vB;
      }
    }
  }
}

// ---------------------------------------------------------------------------
extern "C" void kernel_launch(void* const* d_in, const int* in_sizes, int n_in,
                              void* d_out, int out_size, void* d_ws, size_t ws_size,
                              hipStream_t stream) {
  (void)in_sizes; (void)n_in; (void)out_size; (void)ws_size;
  const float* fs = (const float*)d_in[0];
  float* out = (float*)d_out;
  float* cgA = (float*)d_ws;   // sum_tuples ksteps*64 floats = 91,840 floats (~367 KB)

  // deterministic: CG tables rebuilt every call, fully overwriting the ws region
  cg_init_kernel<<<dim3(NTUP), dim3(256), 0, stream>>>(cgA);
  cg_wmma_kernel<<<dim3(NTUP * 32), dim3(256), 0, stream>>>(fs, out, cgA);
}